// NeuralAdaptation_28613072126392
// MI455X (gfx1250) — compile-verified
//
#include <hip/hip_runtime.h>
#include <cstdint>
#include <cstddef>

// ---------------- problem constants ----------------
constexpr int   Tn  = 20000;          // time steps per row
constexpr int   Rn  = 32 * 40;        // rows = BATCH*CHANNELS = 1280
constexpr int   Ln  = 160;            // segment length (steps)
constexpr int   Sn  = Tn / Ln;        // 125 segments per row
constexpr int   CH  = 32;             // chunk (time steps per tile)
constexpr int   NCH = Ln / CH;        // 5 chunks per segment
constexpr int   WPR = (Sn + 31) / 32; // 4 waves per row (last ragged)
constexpr int   PIN  = 34;            // in-tile pitch: 8B-aligned rows, conflict-free
constexpr int   POUT = 33;            // out-tile pitch: conflict-free for b32 transpose

// ---------------- model constants (fp64 folded to fp32) ----------------
constexpr double dSR = 20000.0, dVI = 0.0005, dVL = 0.005, dPG = 0.03,
                 dPL = 0.06, dPIREST = 0.012, dPIMAX = 0.6, dSPONT = 50.0;
constexpr double dSPER = 1.0 / dSR;
constexpr double dAI = dSPER / dVI;                      // 0.1
constexpr double dAL = dSPER / dVL;                      // 0.01
constexpr double dCI0 = dSPONT / dPIREST;                // 4166.666...
constexpr double dCL0 = dCI0 * (dPIREST + dPL) / dPL;    // 5000
constexpr double dCG  = dCL0 * (1.0 + dPL / dPG) - dCI0 * (dPL / dPG); // 6666.66
constexpr double dLN2 = 0.6931471805599453;
// p1 = log(exp(ln2*PIMAX/PIREST)-1) ~= ln2*50 (correction ~1e-15, below fp32 ulp)
constexpr double dP1  = dLN2 * (dPIMAX / dPIREST);
constexpr double dK2  = dPIREST / dLN2;                  // p3/p1

__device__ __constant__ const float fP1   = (float)dP1;
__device__ __constant__ const float fK2   = (float)dK2;
__device__ __constant__ const float fAI   = (float)dAI;
__device__ __constant__ const float fAL   = (float)dAL;
__device__ __constant__ const float fPL   = (float)dPL;
__device__ __constant__ const float fPG   = (float)dPG;
__device__ __constant__ const float fCG   = (float)dCG;
__device__ __constant__ const float fCI0  = (float)dCI0;
__device__ __constant__ const float fCL0  = (float)dCL0;
__device__ __constant__ const float fSPONT= (float)dSPONT;
__device__ __constant__ const float fB    = (float)(dAI * dPL);            // 0.006
__device__ __constant__ const float fU    = (float)(dAL * dPL);            // 6e-4
__device__ __constant__ const float fG    = (float)(1.0 - dAL*(dPL+dPG));  // 0.9991
__device__ __constant__ const float fC    = (float)(dAL * dPG * dCG);      // 2.0

// ---------------- gfx1250 async-to-LDS helpers ----------------
template <int N>
static __device__ __forceinline__ void waitAsync() {
  asm volatile("s_wait_asynccnt %0" :: "n"(N) : "memory");
}

// GVS form: copies 8 bytes from saddr (64-bit base) + vaddr (32-bit byte offset) to the LDS address in vdst
static __device__ __forceinline__ void asyncLoadB64(uint32_t lds_byte_addr,
                                                    uint32_t byte_off,
                                                    const void* base) {
  asm volatile("global_load_async_to_lds_b64 %0, %1, %2"
               :: "v"(lds_byte_addr), "v"(byte_off), "s"(base)
               : "memory");
}

static __device__ __forceinline__ float softplus_ppi(float x) {
  return fK2 * log1pf(__expf(fP1 * x));
}

// Issue one 32x32 tile as 16 async B64 loads. Instruction m: lanes 0-15 fetch
// the 128B chunk of segment (seg0+2m), lanes 16-31 that of segment (seg0+2m+1),
// transposed into LDS rows 2m / 2m+1 (pitch 34 -> 8B-aligned, conflict-free).
static __device__ __forceinline__ void issueChunk(const float* src, size_t rowbase,
                                                  int seg0, int tc, int lane,
                                                  float* ldsbuf) {
  const int half = lane >> 4;         // 0 or 1: which of the two rows
  const int col  = (lane & 15) * 2;   // even float column within the row
#pragma unroll
  for (int m = 0; m < 16; ++m) {
    int jj = 2 * m + half;
    int sj = seg0 + jj;
    if (sj > Sn - 1) sj = Sn - 1;     // ragged last wave: clamp, stores guarded
    uint32_t off   = (uint32_t)((rowbase + (size_t)sj * Ln + tc + col) * sizeof(float));
    uint32_t laddr = (uint32_t)(uintptr_t)(ldsbuf + jj * PIN + col);
    asyncLoadB64(laddr, off, src);
  }
}

// =============== Phase A: per-segment affine transform composition ===========
__global__ __launch_bounds__(32)
void na_phaseA(const float* __restrict__ ihcl, float* __restrict__ trans) {
  __shared__ float tin[2][CH * PIN];

  const int bid  = blockIdx.x;
  const int row  = bid / WPR;
  const int w    = bid % WPR;
  const int lane = threadIdx.x;
  const int seg0 = w * 32;
  const int segRaw = seg0 + lane;
  const bool valid = segRaw < Sn;
  const int seg = valid ? segRaw : (Sn - 1);
  const size_t rowbase = (size_t)row * Tn;

  // state transform for this segment: out = P*in + q ; start at identity
  float P00 = 1.f, P01 = 0.f, P10 = 0.f, P11 = 1.f, q0 = 0.f, q1 = 0.f;

  issueChunk(ihcl, rowbase, seg0, 0, lane, tin[0]);

  for (int c = 0; c < NCH; ++c) {
    const int cb = c & 1;
    if (c + 1 < NCH) {
      issueChunk(ihcl, rowbase, seg0, (c + 1) * CH, lane, tin[(c + 1) & 1]);
      waitAsync<16>();   // in-order completion: current buffer's 16 are done
    } else {
      waitAsync<0>();
    }
    __syncthreads();     // single-wave WG: barrier ~ S_NOP, keeps LDS ordering

    const int tc = c * CH;
#pragma unroll
    for (int i = 0; i < CH; ++i) {
      float x = tin[cb][lane * PIN + i];
      int k = seg * Ln + tc + i;
      if (k != 0) {  // k==0 contributes identity (scan starts at PPI[1])
        float ppi = softplus_ppi(x);
        float A = 1.0f - fAI * (ppi + fPL);
        float n00 = fmaf(A, P00, fB * P10);
        float n01 = fmaf(A, P01, fB * P11);
        float nq0 = fmaf(A, q0,  fB * q1);
        P10 = fmaf(fU, n00, fG * P10);
        P11 = fmaf(fU, n01, fG * P11);
        q1  = fmaf(fU, nq0, fmaf(fG, q1, fC));
        P00 = n00; P01 = n01; q0 = nq0;
      }
    }
    __syncthreads();
  }

  if (valid) {
    size_t o = ((size_t)row * Sn + segRaw) * 6;
    trans[o + 0] = P00; trans[o + 1] = P01;
    trans[o + 2] = P10; trans[o + 3] = P11;
    trans[o + 4] = q0;  trans[o + 5] = q1;
  }
}

// =============== Phase B: sequential chain of segment transforms =============
__global__ __launch_bounds__(256)
void na_phaseB(const float* __restrict__ trans, float* __restrict__ states) {
  int r = blockIdx.x * blockDim.x + threadIdx.x;
  if (r >= Rn) return;
  float CI = fCI0, CL = fCL0;
  size_t base = (size_t)r * Sn;
  states[base * 2 + 0] = CI;
  states[base * 2 + 1] = CL;
  for (int s = 0; s < Sn - 1; ++s) {
    const float* t = &trans[(base + s) * 6];
    float nCI = fmaf(t[0], CI, fmaf(t[1], CL, t[4]));
    float nCL = fmaf(t[2], CI, fmaf(t[3], CL, t[5]));
    CI = nCI; CL = nCL;
    states[(base + s + 1) * 2 + 0] = CI;
    states[(base + s + 1) * 2 + 1] = CL;
  }
}

// =============== Phase C: replay segments, emit IFR ==========================
__global__ __launch_bounds__(32)
void na_phaseC(const float* __restrict__ ihcl, const float* __restrict__ states,
               float* __restrict__ out) {
  __shared__ float tin[2][CH * PIN];
  __shared__ float tout[CH * POUT];

  const int bid  = blockIdx.x;
  const int row  = bid / WPR;
  const int w    = bid % WPR;
  const int lane = threadIdx.x;
  const int seg0 = w * 32;
  const int segRaw = seg0 + lane;
  const int seg = (segRaw < Sn) ? segRaw : (Sn - 1);
  const size_t rowbase = (size_t)row * Tn;

  float CI = states[((size_t)row * Sn + seg) * 2 + 0];
  float CL = states[((size_t)row * Sn + seg) * 2 + 1];

  issueChunk(ihcl, rowbase, seg0, 0, lane, tin[0]);

  for (int c = 0; c < NCH; ++c) {
    const int cb = c & 1;
    if (c + 1 < NCH) {
      issueChunk(ihcl, rowbase, seg0, (c + 1) * CH, lane, tin[(c + 1) & 1]);
      waitAsync<16>();
    } else {
      waitAsync<0>();
    }
    __syncthreads();

    const int tc = c * CH;
#pragma unroll
    for (int i = 0; i < CH; ++i) {
      float x = tin[cb][lane * PIN + i];
      int k = seg * Ln + tc + i;
      float o;
      if (k == 0) {
        o = fSPONT;  // ifr[0] = spont, no state update
      } else {
        float ppi = softplus_ppi(x);
        CI = CI + fAI * (fPL * (CL - CI) - ppi * CI);
        CL = CL + fAL * (fPG * (fCG - CL) - fPL * (CL - CI)); // uses updated CI
        o = CI * ppi;
      }
      tout[lane * POUT + i] = o;
    }
    __syncthreads();

    // transposed, coalesced stores: lane l writes time (tc+l) of segment seg0+j
#pragma unroll
    for (int j = 0; j < 32; ++j) {
      int sj = seg0 + j;
      if (sj < Sn) {
        out[rowbase + (size_t)sj * Ln + tc + lane] = tout[j * POUT + lane];
      }
    }
    __syncthreads();
  }
}

// ============================ launcher =======================================
extern "C" void kernel_launch(void* const* d_in, const int* in_sizes, int n_in,
                              void* d_out, int out_size, void* d_ws, size_t ws_size,
                              hipStream_t stream) {
  (void)in_sizes; (void)n_in; (void)out_size; (void)ws_size;
  const float* ihcl = (const float*)d_in[0];
  float* out = (float*)d_out;

  float* trans  = (float*)d_ws;                 // Rn*Sn*6 floats (~3.84 MB)
  float* states = trans + (size_t)Rn * Sn * 6;  // Rn*Sn*2 floats (~1.28 MB)

  na_phaseA<<<Rn * WPR, 32, 0, stream>>>(ihcl, trans);
  na_phaseB<<<(Rn + 255) / 256, 256, 0, stream>>>(trans, states);
  na_phaseC<<<Rn * WPR, 32, 0, stream>>>(ihcl, states, out);
}